// TaskCompletionEncoder_78237124264596
// MI455X (gfx1250) — compile-verified
//
#include <hip/hip_runtime.h>
#include <hip/hip_bf16.h>
#include <math.h>

#define D_MODEL 128
#define HIDDEN  256
#define GATES   1024   // 4*HIDDEN
#define K_TPL   20
#define T_LEN   30
#define WIN     20
#define STRIDE  10
#define HEADS   8
#define EDIM    512    // 2*HIDDEN
#define HD      64
#define BATCH   8
#define SEQ     512
#define NSUB    50     // (SEQ-WIN)/STRIDE + 1
#define INFV    1e30f

typedef float v2f __attribute__((ext_vector_type(2)));
typedef float v8f __attribute__((ext_vector_type(8)));

#define WMMA4(a, b, c) \
  __builtin_amdgcn_wmma_f32_16x16x4_f32(false, (a), false, (b), (short)0, (c), false, false)

#if defined(__HIP_DEVICE_COMPILE__) && __has_builtin(__builtin_amdgcn_global_load_async_to_lds_b128)
#define ASYNC_LDS_COPY 1
typedef int v4i_vs __attribute__((vector_size(4 * sizeof(int))));
typedef __attribute__((address_space(1))) v4i_vs* async_gptr;
typedef __attribute__((address_space(3))) v4i_vs* async_lptr;
#endif

__device__ __forceinline__ float sigmoidf_(float x) { return 1.0f / (1.0f + expf(-x)); }

// ---------------------------------------------------------------------------
// fp32 WMMA GEMM: C[M,N] = A[M,K] @ W[N,K]^T + bias[N]
// One wave per 32(M) x 64(N) macro-tile: 2 A-fragments x 4 B-fragments feed
// 8 v_wmma_f32_16x16x4_f32 per K-step from 6 float2 loads (A reused 4x, B 2x).
// Requires N % 64 == 0 (true for all call sites); M is clamp/guarded.
// ---------------------------------------------------------------------------
__global__ void gemm_wmma_f32(const float* __restrict__ A,
                              const float* __restrict__ W,
                              const float* __restrict__ bias,
                              float* __restrict__ C,
                              int M, int N, int K) {
  int lane = threadIdx.x;
  int half = lane >> 4;   // 0 or 1 -> K pair {0,1} vs {2,3}
  int l16  = lane & 15;
  int n0 = blockIdx.x * 64;
  int m0 = blockIdx.y * 32;

  int ar0 = m0 + l16;      if (ar0 >= M) ar0 = M - 1;  // clamp loads; stores guarded
  int ar1 = m0 + 16 + l16; if (ar1 >= M) ar1 = M - 1;

  const float* Ap0 = A + (size_t)ar0 * K + 2 * half;
  const float* Ap1 = A + (size_t)ar1 * K + 2 * half;
  const float* Wp0 = W + (size_t)(n0 + l16) * K + 2 * half;
  const float* Wp1 = Wp0 + (size_t)16 * K;
  const float* Wp2 = Wp0 + (size_t)32 * K;
  const float* Wp3 = Wp0 + (size_t)48 * K;

  v8f z = {0.f, 0.f, 0.f, 0.f, 0.f, 0.f, 0.f, 0.f};
  v8f acc[2][4];
#pragma unroll
  for (int mi = 0; mi < 2; ++mi)
#pragma unroll
    for (int j = 0; j < 4; ++j) acc[mi][j] = z;

  for (int k = 0; k < K; k += 4) {
    v2f a0 = *(const v2f*)(Ap0 + k);
    v2f a1 = *(const v2f*)(Ap1 + k);
    v2f b0 = *(const v2f*)(Wp0 + k);
    v2f b1 = *(const v2f*)(Wp1 + k);
    v2f b2 = *(const v2f*)(Wp2 + k);
    v2f b3 = *(const v2f*)(Wp3 + k);
    __builtin_prefetch(Ap0 + k + 32, 0, 3);
    acc[0][0] = WMMA4(a0, b0, acc[0][0]);
    acc[0][1] = WMMA4(a0, b1, acc[0][1]);
    acc[0][2] = WMMA4(a0, b2, acc[0][2]);
    acc[0][3] = WMMA4(a0, b3, acc[0][3]);
    acc[1][0] = WMMA4(a1, b0, acc[1][0]);
    acc[1][1] = WMMA4(a1, b1, acc[1][1]);
    acc[1][2] = WMMA4(a1, b2, acc[1][2]);
    acc[1][3] = WMMA4(a1, b3, acc[1][3]);
  }

#pragma unroll
  for (int j = 0; j < 4; ++j) {
    int col = n0 + 16 * j + l16;          // N % 64 == 0 -> always in range
    float bval = bias ? bias[col] : 0.0f;
#pragma unroll
    for (int mi = 0; mi < 2; ++mi) {
#pragma unroll
      for (int r = 0; r < 8; ++r) {
        int row = m0 + 16 * mi + r + 8 * half;   // D: VGPR r = rows r / r+8
        if (row < M) C[(size_t)row * N + col] = acc[mi][j][r] + bval;
      }
    }
  }
}

// ---------------------------------------------------------------------------
// Embedding lookup
// ---------------------------------------------------------------------------
__global__ void embed_kernel(const int* __restrict__ ids,
                             const float* __restrict__ emb,
                             float* __restrict__ out) {
  int bs = blockIdx.x;
  int d  = threadIdx.x;
  int id = ids[bs];
  out[(size_t)bs * D_MODEL + d] = emb[(size_t)id * D_MODEL + d];
}

// ---------------------------------------------------------------------------
// biLSTM layer, blockIdx.x = direction.  The per-step recurrent GEMM
// (8x256)@(256x1024) runs on the matrix units: h lives in a zero-padded
// 16x256 LDS tile (rows 8..15 stay zero), each of the 32 waves owns two
// 16-wide N-tiles and issues 2 WMMAs per K-step (64 steps).  Gate math +
// state update follow in a second phase.  LDS: 16KB h + 8KB c + 32KB g.
// ---------------------------------------------------------------------------
__global__ void lstm_layer_kernel(const float* __restrict__ xg_f,
                                  const float* __restrict__ xg_b,
                                  const float* __restrict__ whh_f,
                                  const float* __restrict__ whh_b,
                                  float* __restrict__ out) {
  __shared__ float hpad[16 * HIDDEN];      // [m][kk], rows 8..15 remain zero
  __shared__ float cbuf[BATCH * HIDDEN];   // [b][idx]
  __shared__ float gbuf[BATCH * GATES];    // [b][j]  (h @ whh^T only)

  int dir = blockIdx.x;
  const float* xg  = dir ? xg_b  : xg_f;
  const float* whh = dir ? whh_b : whh_f;
  int tid  = threadIdx.x;          // 1024 threads = 32 waves
  int wave = tid >> 5;
  int lane = tid & 31;
  int half = lane >> 4;
  int l16  = lane & 15;

  for (int i = tid; i < 16 * HIDDEN; i += 1024) hpad[i] = 0.f;
  for (int i = tid; i < BATCH * HIDDEN; i += 1024) cbuf[i] = 0.f;
  __syncthreads();

  const float* ap  = &hpad[l16 * HIDDEN + 2 * half];
  int ncol0 = wave * 32 + l16;                     // two N-tiles per wave
  const float* w0  = whh + (size_t)ncol0 * HIDDEN + 2 * half;
  const float* w1  = w0 + (size_t)16 * HIDDEN;
  int idx = tid & (HIDDEN - 1);
  int bg  = tid >> 8;
  v8f z = {0.f, 0.f, 0.f, 0.f, 0.f, 0.f, 0.f, 0.f};

  for (int step = 0; step < SEQ; ++step) {
    int t = dir ? (SEQ - 1 - step) : step;

    // ---- recurrent matmul on WMMA ----
    v8f acc0 = z, acc1 = z;
    for (int k = 0; k < HIDDEN; k += 4) {
      v2f av = *(const v2f*)(ap + k);      // ds_load_b64 from hpad
      v2f b0 = *(const v2f*)(w0 + k);
      v2f b1 = *(const v2f*)(w1 + k);
      acc0 = WMMA4(av, b0, acc0);
      acc1 = WMMA4(av, b1, acc1);
    }
    if (half == 0) {                        // rows 0..7 live in lanes 0..15
#pragma unroll
      for (int r = 0; r < 8; ++r) {
        gbuf[r * GATES + ncol0]      = acc0[r];
        gbuf[r * GATES + ncol0 + 16] = acc1[r];
      }
    }
    __syncthreads();

    // ---- gates + state update (adds precomputed x @ w_ih^T + b) ----
#pragma unroll
    for (int bb = 0; bb < 2; ++bb) {
      int b = bg * 2 + bb;
      size_t xbase = ((size_t)b * SEQ + t) * GATES;
      float gi = gbuf[b * GATES + idx]              + xg[xbase + idx];
      float gf = gbuf[b * GATES + HIDDEN + idx]     + xg[xbase + HIDDEN + idx];
      float gg = gbuf[b * GATES + 2 * HIDDEN + idx] + xg[xbase + 2 * HIDDEN + idx];
      float go = gbuf[b * GATES + 3 * HIDDEN + idx] + xg[xbase + 3 * HIDDEN + idx];
      float i_ = sigmoidf_(gi), f_ = sigmoidf_(gf), o_ = sigmoidf_(go);
      float g_ = tanhf(gg);
      float c = f_ * cbuf[b * HIDDEN + idx] + i_ * g_;
      float h = o_ * tanhf(c);
      cbuf[b * HIDDEN + idx] = c;
      hpad[b * HIDDEN + idx] = h;
      out[((size_t)b * SEQ + t) * (2 * HIDDEN) + dir * HIDDEN + idx] = h;
    }
    __syncthreads();
  }
}

// ---------------------------------------------------------------------------
// Fused attention: A_red[b,k,h,t] = sum_s softmax_t(q . k / 8).
// K-tile staged into LDS via gfx1250 async global->LDS copy when available.
// ---------------------------------------------------------------------------
__global__ void attn_reduce_kernel(const float* __restrict__ q,
                                   const float* __restrict__ kk,
                                   float* __restrict__ A_red) {
  __shared__ float kl[T_LEN * HD];   // 7.5 KB
  __shared__ float Aacc[T_LEN];

  int h = blockIdx.x % HEADS;
  int k = (blockIdx.x / HEADS) % K_TPL;
  int b = blockIdx.x / (HEADS * K_TPL);
  int tid = threadIdx.x;   // 256 threads

#ifdef ASYNC_LDS_COPY
  for (int i = tid; i < (T_LEN * HD) / 4; i += 256) {
    int t  = i / (HD / 4);
    int dq = (i % (HD / 4)) * 4;
    const float* src = kk + (size_t)(k * T_LEN + t) * EDIM + h * HD + dq;
    __builtin_amdgcn_global_load_async_to_lds_b128(
        (async_gptr)src, (async_lptr)&kl[t * HD + dq], 0, 0);
  }
  if (tid < T_LEN) Aacc[tid] = 0.f;
  asm volatile("s_wait_asynccnt 0x0" ::: "memory");
  __syncthreads();
#else
  for (int i = tid; i < T_LEN * HD; i += 256) {
    int t = i / HD, d = i % HD;
    kl[t * HD + d] = kk[((size_t)(k * T_LEN + t)) * EDIM + h * HD + d];
  }
  if (tid < T_LEN) Aacc[tid] = 0.f;
  __syncthreads();
#endif

  for (int s = tid; s < SEQ; s += 256) {
    float sc[T_LEN];
#pragma unroll
    for (int t = 0; t < T_LEN; ++t) sc[t] = 0.f;
    const float* qrow = q + ((size_t)(b * SEQ + s)) * EDIM + h * HD;
    for (int d = 0; d < HD; ++d) {
      float qd = qrow[d];
#pragma unroll
      for (int t = 0; t < T_LEN; ++t) sc[t] += qd * kl[t * HD + d];
    }
    float mx = -INFV;
#pragma unroll
    for (int t = 0; t < T_LEN; ++t) { sc[t] *= 0.125f; mx = fmaxf(mx, sc[t]); }
    float sum = 0.f;
#pragma unroll
    for (int t = 0; t < T_LEN; ++t) { sc[t] = expf(sc[t] - mx); sum += sc[t]; }
    float inv = 1.0f / sum;
#pragma unroll
    for (int t = 0; t < T_LEN; ++t) atomicAdd(&Aacc[t], sc[t] * inv);
  }
  __syncthreads();
  if (tid < T_LEN)
    A_red[(((size_t)b * K_TPL + k) * HEADS + h) * T_LEN + tid] = Aacc[tid];
}

// ---------------------------------------------------------------------------
// h_attn_pre[b,col] = (1/(K*S)) * sum_{k,t} A_red[b,k,h,t] * v[k*30+t, col]
// ---------------------------------------------------------------------------
__global__ void attn_combine_kernel(const float* __restrict__ A_red,
                                    const float* __restrict__ v,
                                    float* __restrict__ h_attn_pre) {
  int b = blockIdx.x;
  int col = threadIdx.x;
  int h = col >> 6;
  float acc = 0.f;
  for (int k = 0; k < K_TPL; ++k) {
    const float* ar = A_red + (((size_t)b * K_TPL + k) * HEADS + h) * T_LEN;
    for (int t = 0; t < T_LEN; ++t)
      acc += ar[t] * v[((size_t)(k * T_LEN + t)) * EDIM + col];
  }
  h_attn_pre[(size_t)b * EDIM + col] = acc * (1.0f / (float)(K_TPL * SEQ));
}

// ---------------------------------------------------------------------------
// DTW: one block per (subsequence, template).
// ---------------------------------------------------------------------------
__global__ void dtw_kernel(const float* __restrict__ emb_seq,
                           const float* __restrict__ templates,
                           float* __restrict__ dists) {
  __shared__ float cost[WIN * T_LEN];   // 600 floats
  int bsub = blockIdx.x;                // 0..399
  int tk   = blockIdx.y;                // 0..19
  int b    = bsub / NSUB;
  int start = (bsub % NSUB) * STRIDE;
  int tid = threadIdx.x;                // 128 threads

  for (int e = tid; e < WIN * T_LEN; e += 128) {
    int i = e / T_LEN, j = e % T_LEN;
    const float* srow = emb_seq + ((size_t)b * SEQ + start + i) * D_MODEL;
    const float* trow = templates + ((size_t)tk * T_LEN + j) * D_MODEL;
    float d2 = 0.f;
    for (int d = 0; d < D_MODEL; ++d) {
      float df = srow[d] - trow[d];
      d2 += df * df;
    }
    cost[e] = sqrtf(fmaxf(d2, 0.f));
  }
  __syncthreads();

  if (tid == 0) {
    float prev[T_LEN], cur[T_LEN];
    prev[0] = cost[0];
#pragma unroll
    for (int j = 1; j < T_LEN; ++j) prev[j] = prev[j - 1] + cost[j];
    for (int i = 1; i < WIN; ++i) {
      const float* cr = &cost[i * T_LEN];
      cur[0] = cr[0] + prev[0];
#pragma unroll
      for (int j = 1; j < T_LEN; ++j)
        cur[j] = cr[j] + fminf(cur[j - 1], fminf(prev[j], prev[j - 1]));
#pragma unroll
      for (int j = 0; j < T_LEN; ++j) prev[j] = cur[j];
    }
    dists[(size_t)bsub * K_TPL + tk] = prev[T_LEN - 1];
  }
}

// ---------------------------------------------------------------------------
// tir[b] = mean_sub( max_k exp(-dist/tau) < theta )
// ---------------------------------------------------------------------------
__global__ void tir_kernel(const float* __restrict__ dists,
                           const float* __restrict__ theta,
                           const float* __restrict__ tau,
                           float* __restrict__ tir_buf,
                           float* __restrict__ out_tir) {
  __shared__ float cnt[64];
  int b = blockIdx.x;
  int tid = threadIdx.x;
  float th = theta[0], itau = 1.0f / tau[0];
  float c = 0.f;
  for (int sub = tid; sub < NSUB; sub += 64) {
    float mx = -INFV;
    const float* dr = dists + ((size_t)b * NSUB + sub) * K_TPL;
    for (int k = 0; k < K_TPL; ++k) mx = fmaxf(mx, expf(-dr[k] * itau));
    c += (mx < th) ? 1.0f : 0.0f;
  }
  cnt[tid] = c;
  __syncthreads();
  for (int off = 32; off > 0; off >>= 1) {
    if (tid < off) cnt[tid] += cnt[tid + off];
    __syncthreads();
  }
  if (tid == 0) {
    float t = cnt[0] * (1.0f / (float)NSUB);
    tir_buf[b] = t;
    out_tir[b] = t;
  }
}

// ---------------------------------------------------------------------------
// concat[b, 0:512] = h_attn ; concat[b, 512:544] = tir[b]*tir_w + tir_b
// ---------------------------------------------------------------------------
__global__ void concat_kernel(const float* __restrict__ h_attn,
                              const float* __restrict__ tir_buf,
                              const float* __restrict__ tir_w,
                              const float* __restrict__ tir_b,
                              float* __restrict__ concat) {
  int b = blockIdx.x;
  int j = threadIdx.x;   // 0..543
  float v;
  if (j < EDIM) v = h_attn[(size_t)b * EDIM + j];
  else          v = tir_buf[b] * tir_w[j - EDIM] + tir_b[j - EDIM];
  concat[(size_t)b * (EDIM + 32) + j] = v;
}

// ---------------------------------------------------------------------------
extern "C" void kernel_launch(void* const* d_in, const int* in_sizes, int n_in,
                              void* d_out, int out_size, void* d_ws, size_t ws_size,
                              hipStream_t stream) {
  (void)in_sizes; (void)n_in; (void)out_size; (void)ws_size;

  const int*   ids       = (const int*)  d_in[0];
  const float* emb       = (const float*)d_in[1];
  const float* l0f_wih   = (const float*)d_in[2];
  const float* l0f_whh   = (const float*)d_in[3];
  const float* l0f_b     = (const float*)d_in[4];
  const float* l0b_wih   = (const float*)d_in[5];
  const float* l0b_whh   = (const float*)d_in[6];
  const float* l0b_b     = (const float*)d_in[7];
  const float* l1f_wih   = (const float*)d_in[8];
  const float* l1f_whh   = (const float*)d_in[9];
  const float* l1f_b     = (const float*)d_in[10];
  const float* l1b_wih   = (const float*)d_in[11];
  const float* l1b_whh   = (const float*)d_in[12];
  const float* l1b_b     = (const float*)d_in[13];
  const float* templates = (const float*)d_in[14];
  const float* tproj_w   = (const float*)d_in[15];
  const float* tproj_b   = (const float*)d_in[16];
  const float* inproj_w  = (const float*)d_in[17];
  const float* inproj_b  = (const float*)d_in[18];
  const float* outproj_w = (const float*)d_in[19];
  const float* outproj_b = (const float*)d_in[20];
  const float* tir_w     = (const float*)d_in[21];
  const float* tir_b     = (const float*)d_in[22];
  const float* o_w       = (const float*)d_in[23];
  const float* o_b       = (const float*)d_in[24];
  const float* theta     = (const float*)d_in[25];
  const float* tau       = (const float*)d_in[26];

  float* ws = (float*)d_ws;
  const int BS = BATCH * SEQ;                 // 4096
  size_t off = 0;
  float* emb_seq    = ws + off; off += (size_t)BS * D_MODEL;
  float* xgf        = ws + off; off += (size_t)BS * GATES;
  float* xgb        = ws + off; off += (size_t)BS * GATES;
  float* h_l0       = ws + off; off += (size_t)BS * EDIM;
  float* h_l1       = ws + off; off += (size_t)BS * EDIM;
  float* tpl        = ws + off; off += (size_t)(K_TPL * T_LEN) * EDIM;
  float* qbuf       = ws + off; off += (size_t)BS * EDIM;
  float* kbuf       = ws + off; off += (size_t)(K_TPL * T_LEN) * EDIM;
  float* vbuf       = ws + off; off += (size_t)(K_TPL * T_LEN) * EDIM;
  float* A_red      = ws + off; off += (size_t)BATCH * K_TPL * HEADS * T_LEN;
  float* h_attn_pre = ws + off; off += (size_t)BATCH * EDIM;
  float* h_attn     = ws + off; off += (size_t)BATCH * EDIM;
  float* dists      = ws + off; off += (size_t)BATCH * NSUB * K_TPL;
  float* tir_buf    = ws + off; off += BATCH;
  float* concat     = ws + off; off += (size_t)BATCH * (EDIM + 32);

  float* out_htc = (float*)d_out;               // 8x128
  float* out_tir = (float*)d_out + BATCH * D_MODEL;

  auto gemm = [&](const float* A, const float* W, const float* bias, float* C,
                  int M, int N, int K) {
    dim3 g(N / 64, (M + 31) / 32);
    gemm_wmma_f32<<<g, 32, 0, stream>>>(A, W, bias, C, M, N, K);
  };

  // 1) embedding
  embed_kernel<<<BS, D_MODEL, 0, stream>>>(ids, emb, emb_seq);

  // 2) layer 0
  gemm(emb_seq, l0f_wih, l0f_b, xgf, BS, GATES, D_MODEL);
  gemm(emb_seq, l0b_wih, l0b_b, xgb, BS, GATES, D_MODEL);
  lstm_layer_kernel<<<2, 1024, 0, stream>>>(xgf, xgb, l0f_whh, l0b_whh, h_l0);

  // 3) layer 1
  gemm(h_l0, l1f_wih, l1f_b, xgf, BS, GATES, EDIM);
  gemm(h_l0, l1b_wih, l1b_b, xgb, BS, GATES, EDIM);
  lstm_layer_kernel<<<2, 1024, 0, stream>>>(xgf, xgb, l1f_whh, l1b_whh, h_l1);

  // 4) template projection + Q/K/V
  gemm(templates, tproj_w, tproj_b, tpl, K_TPL * T_LEN, EDIM, D_MODEL);
  gemm(h_l1, inproj_w,                           inproj_b,            qbuf, BS, EDIM, EDIM);
  gemm(tpl,  inproj_w + (size_t)EDIM * EDIM,     inproj_b + EDIM,     kbuf, K_TPL * T_LEN, EDIM, EDIM);
  gemm(tpl,  inproj_w + (size_t)2 * EDIM * EDIM, inproj_b + 2 * EDIM, vbuf, K_TPL * T_LEN, EDIM, EDIM);

  // 5) attention
  attn_reduce_kernel<<<BATCH * K_TPL * HEADS, 256, 0, stream>>>(qbuf, kbuf, A_red);
  attn_combine_kernel<<<BATCH, EDIM, 0, stream>>>(A_red, vbuf, h_attn_pre);
  gemm(h_attn_pre, outproj_w, outproj_b, h_attn, BATCH, EDIM, EDIM);

  // 6) DTW branch
  dtw_kernel<<<dim3(BATCH * NSUB, K_TPL), 128, 0, stream>>>(emb_seq, templates, dists);
  tir_kernel<<<BATCH, 64, 0, stream>>>(dists, theta, tau, tir_buf, out_tir);

  // 7) concat + final head
  concat_kernel<<<BATCH, EDIM + 32, 0, stream>>>(h_attn, tir_buf, tir_w, tir_b, concat);
  gemm(concat, o_w, o_b, out_htc, BATCH, D_MODEL, EDIM + 32);
}